// KirchhoffVoltageLaw_38010460570137
// MI455X (gfx1250) — compile-verified
//
#include <hip/hip_runtime.h>
#include <math.h>

// ---------------------------------------------------------------------------
// KirchhoffVoltageLaw reduction for MI455X (gfx1250, wave32).
//
// Pass 1: grid-stride over edges, 11 f32 accumulators per thread:
//   [0]=W=sum(w)  [1..4]=S_j=sum(w*p_j)  [5..8]=A_j=sum(w*p_j^2)
//   [9]=D=sum(drop)  [10]=D2=sum(drop^2)
// Intra-wave: shfl_xor butterfly (wave32). Cross-wave (8 waves/block):
// two chained V_WMMA_F32_16X16X4_F32 against an all-ones B matrix —
// D[m][n] = sum_k A[m][k] is the exact 8-wave sum of value m.
// Pass 2: single block reduces 1024 block partials in f64 and applies the
// closed-form weighted-variance / ddof=1 variance formulas.
// ---------------------------------------------------------------------------

typedef __attribute__((ext_vector_type(2))) float v2f;
typedef __attribute__((ext_vector_type(8))) float v8f;

#define NACC         11
#define PART_STRIDE  16
#define GRID_BLOCKS  1024
#define BLOCK_THREADS 256
#define EPS          1e-6

__global__ __launch_bounds__(BLOCK_THREADS)
void kvl_pass1(const float* __restrict__ node_features,
               const long long* __restrict__ edge_index,
               const float* __restrict__ edge_probs,
               const float* __restrict__ edge_params,
               float* __restrict__ partials,
               long long E)
{
    const long long* __restrict__ srcI = edge_index;
    const long long* __restrict__ dstI = edge_index + E;
    const float4*    __restrict__ prm4 = (const float4*)edge_params;
    const float2*    __restrict__ nf2  = (const float2*)node_features; // row = 2 float2, we use the first

    float W = 0.f;
    float S0 = 0.f, S1 = 0.f, S2 = 0.f, S3 = 0.f;
    float A0 = 0.f, A1 = 0.f, A2 = 0.f, A3 = 0.f;
    float D = 0.f, D2 = 0.f;

    const long long stride = (long long)gridDim.x * (long long)blockDim.x;
    for (long long e = (long long)blockIdx.x * blockDim.x + threadIdx.x; e < E; e += stride) {
        float  w = edge_probs[e];
        float4 p = prm4[e];

        W  += w;
        S0 += w * p.x;        S1 += w * p.y;
        S2 += w * p.z;        S3 += w * p.w;
        A0 += w * p.x * p.x;  A1 += w * p.y * p.y;
        A2 += w * p.z * p.z;  A3 += w * p.w * p.w;

        long long s = srcI[e];
        long long d = dstI[e];
        float2 fs = nf2[2 * s];   // node_features[s, 0:2]
        float2 fd = nf2[2 * d];   // node_features[d, 0:2]
        float dvx = fs.x - fd.x;
        float dvy = fs.y - fd.y;
        float mag = sqrtf(dvx * dvx + dvy * dvy);
        float drop = mag * w;
        D  += drop;
        D2 += drop * drop;
    }

    float acc[NACC] = {W, S0, S1, S2, S3, A0, A1, A2, A3, D, D2};

    // Intra-wave butterfly reduction (wave32): every lane ends with the wave sum.
    #pragma unroll
    for (int off = 16; off > 0; off >>= 1) {
        #pragma unroll
        for (int j = 0; j < NACC; ++j)
            acc[j] += __shfl_xor(acc[j], off, 32);
    }

    __shared__ float wbuf[8][16];   // [wave][value], padded 11 -> 16 with zeros
    const int lane = threadIdx.x & 31;
    const int wave = threadIdx.x >> 5;
    if (lane == 0) {
        #pragma unroll
        for (int j = 0; j < 16; ++j)
            wbuf[wave][j] = (j < NACC) ? acc[j] : 0.0f;
    }
    __syncthreads();

    // Cross-wave reduction with V_WMMA_F32_16X16X4_F32 on wave 0 only
    // (full 32-lane wave => EXEC all-ones, as WMMA requires).
    //
    // A-matrix layout (16x4 f32): lanes 0-15 hold row M=lane with K=0 (vgpr0)
    // and K=1 (vgpr1); lanes 16-31 hold K=2 (vgpr0), K=3 (vgpr1).
    // We set A[m][k] = wavePartial_k[m], B = all ones, so
    // D[m][n] = sum_k A[m][k] = sum over 4 waves of value m (exact).
    // Two chained WMMAs cover all 8 waves.
    if (threadIdx.x < 32) {
        const int row = lane & 15;
        const int hi  = lane >> 4;   // 0: K=0/1 half, 1: K=2/3 half

        v2f a1, a2, b;
        a1.x = wbuf[hi * 2 + 0][row];   // waves 0..3
        a1.y = wbuf[hi * 2 + 1][row];
        a2.x = wbuf[hi * 2 + 4][row];   // waves 4..7
        a2.y = wbuf[hi * 2 + 5][row];
        b.x = 1.0f; b.y = 1.0f;

        v8f c = {};
        c = __builtin_amdgcn_wmma_f32_16x16x4_f32(false, a1, false, b, (short)0, c, false, false);
        c = __builtin_amdgcn_wmma_f32_16x16x4_f32(false, a2, false, b, (short)0, c, false, false);

        // D rows 0..7 live in c[0..7] of lanes 0-15; rows 8..15 in lanes 16-31.
        float* outp = partials + (size_t)blockIdx.x * PART_STRIDE;
        if (lane == 0) {
            #pragma unroll
            for (int j = 0; j < 8; ++j) outp[j] = c[j];          // values 0..7
        } else if (lane == 16) {
            #pragma unroll
            for (int j = 0; j < 3; ++j) outp[8 + j] = c[j];      // values 8..10
        }
    }
}

__global__ __launch_bounds__(BLOCK_THREADS)
void kvl_pass2(const float* __restrict__ partials,
               float* __restrict__ out,
               int nblocks,
               long long E)
{
    double acc[NACC];
    #pragma unroll
    for (int j = 0; j < NACC; ++j) acc[j] = 0.0;

    for (int r = threadIdx.x; r < nblocks; r += BLOCK_THREADS) {
        const float* p = partials + (size_t)r * PART_STRIDE;
        #pragma unroll
        for (int j = 0; j < NACC; ++j) acc[j] += (double)p[j];
    }

    __shared__ double sd[BLOCK_THREADS][NACC + 1];  // +1 to dodge bank conflicts
    #pragma unroll
    for (int j = 0; j < NACC; ++j) sd[threadIdx.x][j] = acc[j];
    __syncthreads();

    for (int s = BLOCK_THREADS / 2; s > 0; s >>= 1) {
        if (threadIdx.x < (unsigned)s) {
            #pragma unroll
            for (int j = 0; j < NACC; ++j)
                sd[threadIdx.x][j] += sd[threadIdx.x + s][j];
        }
        __syncthreads();
    }

    if (threadIdx.x == 0) {
        double W  = sd[0][0];
        double S[4] = {sd[0][1], sd[0][2], sd[0][3], sd[0][4]};
        double A[4] = {sd[0][5], sd[0][6], sd[0][7], sd[0][8]};
        double D  = sd[0][9];
        double D2 = sd[0][10];

        // param_consistency = mean_j [ A_j - 2 m_j S_j + m_j^2 W ],  m_j = S_j/(W+EPS)
        double denom = W + EPS;
        double pc = 0.0;
        #pragma unroll
        for (int j = 0; j < 4; ++j) {
            double m = S[j] / denom;
            pc += A[j] - 2.0 * m * S[j] + m * m * W;
        }
        pc *= 0.25;

        // var(drops, ddof=1) = (D2 - D^2/E) / (E-1)
        double vc = (D2 - (D * D) / (double)E) / (double)(E - 1);

        out[0] = (float)(pc + vc);
    }
}

extern "C" void kernel_launch(void* const* d_in, const int* in_sizes, int n_in,
                              void* d_out, int out_size, void* d_ws, size_t ws_size,
                              hipStream_t stream) {
    const float*     node_features = (const float*)d_in[0];
    const long long* edge_index    = (const long long*)d_in[1];  // int64 (2, E)
    const float*     edge_probs    = (const float*)d_in[2];
    const float*     edge_params   = (const float*)d_in[3];
    const long long  E = (long long)in_sizes[2];

    float* partials = (float*)d_ws;   // GRID_BLOCKS * PART_STRIDE floats = 64 KB

    kvl_pass1<<<GRID_BLOCKS, BLOCK_THREADS, 0, stream>>>(
        node_features, edge_index, edge_probs, edge_params, partials, E);
    kvl_pass2<<<1, BLOCK_THREADS, 0, stream>>>(
        partials, (float*)d_out, GRID_BLOCKS, E);
}